// SinkhornFromScores_84370337563296
// MI455X (gfx1250) — compile-verified
//
#include <hip/hip_runtime.h>

#define B_      8
#define N_      2048
#define M_      2048
#define INV_TAU 10.0f
#define LOG2E   1.4426950408889634f
#define C1      14.426950408889634f     // INV_TAU * LOG2E
#define LOG_AB  (-7.6246189861593985f)  // -log(2048)
#define ITERS   20

typedef float v2f __attribute__((ext_vector_type(2)));
typedef float v8f __attribute__((ext_vector_type(8)));

__device__ __forceinline__ float fexp2(float x) { return __builtin_amdgcn_exp2f(x); }

// ---------------- init helpers ----------------

__global__ void k_zero(float* p) {
    p[blockIdx.x * 256 + threadIdx.x] = 0.0f;
}

// one wave per row: R[row] = max_j(log_alpha[row][j]) * (1/tau)
__global__ void k_rowmax(const float* __restrict__ la, float* __restrict__ R) {
    int wid = threadIdx.x >> 5, lane = threadIdx.x & 31;
    int row = blockIdx.x * 8 + wid;                  // [0, B*N)
    const float4* p4 = (const float4*)(la + (size_t)row * M_);
    float m = -__builtin_inff();
    for (int k = lane; k < M_ / 4; k += 32) {
        float4 v = p4[k];
        m = fmaxf(m, fmaxf(fmaxf(v.x, v.y), fmaxf(v.z, v.w)));
    }
    for (int off = 16; off; off >>= 1) m = fmaxf(m, __shfl_xor(m, off, 32));
    if (lane == 0) R[row] = m * INV_TAU;
}

// thread per column: Cm[b][j] = max_i(log_alpha[b][i][j]) * (1/tau)
__global__ void k_colmax(const float* __restrict__ la, float* __restrict__ Cm) {
    int j = blockIdx.x * 256 + threadIdx.x;
    int b = blockIdx.y;
    const float* p = la + ((size_t)b << 22) + j;
    float m = -__builtin_inff();
    for (int i = 0; i < N_; i += 4) {
        __builtin_prefetch(p + (size_t)(i + 16) * M_, 0, 1);
        float a0 = p[(size_t)(i + 0) * M_];
        float a1 = p[(size_t)(i + 1) * M_];
        float a2 = p[(size_t)(i + 2) * M_];
        float a3 = p[(size_t)(i + 3) * M_];
        m = fmaxf(m, fmaxf(fmaxf(a0, a1), fmaxf(a2, a3)));
    }
    Cm[b * M_ + j] = m * INV_TAU;
}

// block per batch: out[b] = max over 2048-vector
__global__ void k_vmax(const float* __restrict__ v, float* __restrict__ out) {
    __shared__ float s[256];
    int b = blockIdx.x;
    float m = -__builtin_inff();
    for (int t = threadIdx.x; t < M_; t += 256) m = fmaxf(m, v[b * M_ + t]);
    s[threadIdx.x] = m;
    __syncthreads();
    for (int off = 128; off; off >>= 1) {
        if (threadIdx.x < off) s[threadIdx.x] = fmaxf(s[threadIdx.x], s[threadIdx.x + off]);
        __syncthreads();
    }
    if (threadIdx.x == 0) out[b] = s[0];
}

// ---------------- row dual update: f = log_a - lse_j(log_K + g) ----------------
// one wave per row; base-2 domain: summand = exp2(la*C1 + g2_j - shift2)
__global__ void k_rowupdate(const float* __restrict__ la, const float* __restrict__ g2,
                            const float* __restrict__ R, const float* __restrict__ gmax,
                            float* __restrict__ f, float* __restrict__ f2) {
    int wid = threadIdx.x >> 5, lane = threadIdx.x & 31;
    int row = blockIdx.x * 8 + wid;                  // [0, B*N)
    int b = row >> 11;
    const float4* p4 = (const float4*)(la + (size_t)row * M_);
    const float4* g4 = (const float4*)(g2 + b * M_);
    float shift  = R[row] + gmax[b];                 // >= max_j(log_K + g_j)
    float shift2 = shift * LOG2E;
    float acc = 0.0f;
    for (int k = lane; k < M_ / 4; k += 32) {
        float4 v = p4[k];
        float4 gg = g4[k];
        acc += fexp2(fmaf(v.x, C1, gg.x) - shift2);
        acc += fexp2(fmaf(v.y, C1, gg.y) - shift2);
        acc += fexp2(fmaf(v.z, C1, gg.z) - shift2);
        acc += fexp2(fmaf(v.w, C1, gg.w) - shift2);
    }
    for (int off = 16; off; off >>= 1) acc += __shfl_xor(acc, off, 32);
    if (lane == 0) {
        float val = LOG_AB - (shift + __logf(acc));
        f[row]  = val;
        f2[row] = val * LOG2E;
    }
}

// ---------------- col dual update: g = log_b - lse_i(log_K + f) ----------------
// WMMA reduction: D = ones(16x4) x B(4x16) accumulates per-column sums in C.
// Lane layout of B: lanes 0-15 -> rows r0,r0+1; lanes 16-31 -> rows r0+2,r0+3
// at column j0+(lane&15), so each half-wave load is one contiguous 64B line.
__global__ void k_colupdate_wmma(const float* __restrict__ la, const float* __restrict__ f2,
                                 const float* __restrict__ Cm, const float* __restrict__ fmax,
                                 float* __restrict__ g, float* __restrict__ g2) {
    __shared__ float fs2[N_];                        // f * log2e, staged in LDS
    int b = blockIdx.y;
    int tid = threadIdx.x;
    for (int r = tid; r < N_; r += 256) fs2[r] = f2[b * N_ + r];
    __syncthreads();

    int wid = tid >> 5, lane = tid & 31;
    int j = blockIdx.x * 128 + wid * 16 + (lane & 15);   // this lane's column
    int krow = (lane >> 4) << 1;                          // 0 or 2
    float shift  = Cm[b * M_ + j] + fmax[b];              // >= max_i(log_K + f_i)
    float shift2 = shift * LOG2E;
    const float* base = la + ((size_t)b << 22) + j;

    v2f ones; ones.x = 1.0f; ones.y = 1.0f;
    v8f acc = {};
    for (int r0 = 0; r0 < N_; r0 += 4) {
        int r = r0 + krow;
        v2f bb;
        bb.x = fexp2(fmaf(base[(size_t)(r + 0) * M_], C1, fs2[r + 0]) - shift2);
        bb.y = fexp2(fmaf(base[(size_t)(r + 1) * M_], C1, fs2[r + 1]) - shift2);
        // D[m][n] = sum_k 1 * B[k][n] + C : accumulates column sums across K
        acc = __builtin_amdgcn_wmma_f32_16x16x4_f32(
            /*neg_a=*/false, ones, /*neg_b=*/false, bb,
            /*c_mod=*/(short)0, acc, /*reuse_a=*/false, /*reuse_b=*/false);
    }
    // every row of D holds the column sums; VGPR0 lanes 0-15 = N=0..15
    if (lane < 16) {
        float val = LOG_AB - (shift + __logf(acc[0]));
        g[b * M_ + j]  = val;
        g2[b * M_ + j] = val * LOG2E;
    }
}

// ---------------- final: P = exp2(la*C1 + f2_i + g2_j), float4 vectorized ----------------
__global__ void k_final(const float* __restrict__ la, const float* __restrict__ f2,
                        const float* __restrict__ g2, float* __restrict__ out) {
    size_t t = (size_t)blockIdx.x * 256 + threadIdx.x;   // float4 index
    size_t flat = t * 4;
    int b = (int)(flat >> 22);
    int i = (int)((flat >> 11) & (N_ - 1));
    int j = (int)(flat & (M_ - 1));
    float4 v = ((const float4*)la)[t];
    float4 gg = *(const float4*)(g2 + b * M_ + j);
    float fv = f2[b * N_ + i];
    float4 o;
    o.x = fexp2(fmaf(v.x, C1, fv + gg.x));
    o.y = fexp2(fmaf(v.y, C1, fv + gg.y));
    o.z = fexp2(fmaf(v.z, C1, fv + gg.z));
    o.w = fexp2(fmaf(v.w, C1, fv + gg.w));
    ((float4*)out)[t] = o;
}

extern "C" void kernel_launch(void* const* d_in, const int* in_sizes, int n_in,
                              void* d_out, int out_size, void* d_ws, size_t ws_size,
                              hipStream_t stream) {
    const float* la = (const float*)d_in[0];
    float* out = (float*)d_out;
    float* ws = (float*)d_ws;

    float* R    = ws;                 // B*N  = 16384
    float* Cm   = ws + 16384;         // B*M  = 16384
    float* f    = ws + 32768;         // B*N
    float* g    = ws + 49152;         // B*M
    float* f2   = ws + 65536;         // B*N  (f * log2e)
    float* g2   = ws + 81920;         // B*M  (g * log2e)
    float* gmax = ws + 98304;         // B
    float* fmax = ws + 98312;         // B

    // zero g and g2 (contiguous 32768 floats starting at g? no: g at 49152, g2 at 81920)
    k_zero<<<B_ * M_ / 256, 256, 0, stream>>>(g);
    k_zero<<<B_ * M_ / 256, 256, 0, stream>>>(g2);
    k_rowmax<<<B_ * N_ / 8, 256, 0, stream>>>(la, R);
    k_colmax<<<dim3(M_ / 256, B_), 256, 0, stream>>>(la, Cm);

    for (int it = 0; it < ITERS; ++it) {
        k_vmax<<<B_, 256, 0, stream>>>(g, gmax);
        k_rowupdate<<<B_ * N_ / 8, 256, 0, stream>>>(la, g2, R, gmax, f, f2);
        k_vmax<<<B_, 256, 0, stream>>>(f, fmax);
        k_colupdate_wmma<<<dim3(M_ / 128, B_), 256, 0, stream>>>(la, f2, Cm, fmax, g, g2);
    }

    k_final<<<(size_t)B_ * N_ * M_ / 4 / 256, 256, 0, stream>>>(la, f2, g2, out);
}